// PyramidROIAlign_2388001816806
// MI455X (gfx1250) — compile-verified
//
#include <hip/hip_runtime.h>
#include <cmath>

// ---------------------------------------------------------------------------
// PyramidROIAlign for MI455X (gfx1250).
// Bandwidth/gather-bound op: ~0.5 GB traffic, ~0.2 GFLOP. Strategy:
//  - compute ONLY the selected FPN level per box (4x less work than reference)
//  - one wave32 per output pixel; lane owns 8 of 256 channels
//  - corners staged via CDNA5 async global->LDS copies (ASYNCcnt), double
//    buffered so pixel i+1's gather overlaps pixel i's blend+store
//  - non-temporal output stores: fmaps (~178 MB) nearly fill the 192 MB L2;
//    keep the write stream (~100 MB) from evicting the gather working set
// ---------------------------------------------------------------------------

#define WAVES_PER_BLOCK 4
#define BLOCK_THREADS   (WAVES_PER_BLOCK * 32)
#define POOL            7
#define CCH             256   // channels
#define CHUNK           8     // channels per lane (256 / 32)

typedef float v4f __attribute__((ext_vector_type(4)));

// ---- CDNA5 async global->LDS b128 copy (per-lane 16B) ---------------------
typedef int v4i __attribute__((vector_size(4 * sizeof(int))));
typedef __attribute__((address_space(1))) v4i* gv4i_p;  // global (__device__)
typedef __attribute__((address_space(3))) v4i* lv4i_p;  // LDS (__shared__)

#if __has_builtin(__builtin_amdgcn_global_load_async_to_lds_b128)
__device__ __forceinline__ void async_copy_b128(const float* g, float* l) {
  __builtin_amdgcn_global_load_async_to_lds_b128(
      (gv4i_p)(void*)g, (lv4i_p)(void*)l, /*imm offset*/ 0, /*cpol*/ 0);
}
#else
__device__ __forceinline__ void async_copy_b128(const float* g, float* l) {
  __attribute__((address_space(3))) void* lp =
      (__attribute__((address_space(3))) void*)(void*)l;
  asm volatile("global_load_async_to_lds_b128 %0, %1, off"
               :: "v"(lp), "v"(g) : "memory");
}
#endif

#if __has_builtin(__builtin_amdgcn_s_wait_asynccnt)
#define WAIT_ASYNC(n) do { __builtin_amdgcn_s_wait_asynccnt(n); \
                           asm volatile("" ::: "memory"); } while (0)
#else
#define WAIT_ASYNC(n) asm volatile("s_wait_asynccnt %0" :: "n"(n) : "memory")
#endif

__device__ __forceinline__ void store_nt4(float* p, v4f v) {
  __builtin_nontemporal_store(v, (v4f*)p);
}

// ---- bilinear blend of 8 channels from LDS-staged corners, then store -----
__device__ __forceinline__ void blend_store(const float* __restrict__ buf,
                                            float* __restrict__ o,
                                            float lx, float ly, int valid,
                                            int c0) {
  if (!valid) {
    v4f z = {0.f, 0.f, 0.f, 0.f};
    store_nt4(o + c0, z);
    store_nt4(o + c0 + 4, z);
    return;
  }
#pragma unroll
  for (int h = 0; h < 2; ++h) {
    v4f tl = *(const v4f*)(buf +           c0 + 4 * h);
    v4f tr = *(const v4f*)(buf + 1 * CCH + c0 + 4 * h);
    v4f bl = *(const v4f*)(buf + 2 * CCH + c0 + 4 * h);
    v4f br = *(const v4f*)(buf + 3 * CCH + c0 + 4 * h);
    v4f r;
    float t, b;
    t = fmaf(tr.x - tl.x, lx, tl.x); b = fmaf(br.x - bl.x, lx, bl.x);
    r.x = fmaf(b - t, ly, t);
    t = fmaf(tr.y - tl.y, lx, tl.y); b = fmaf(br.y - bl.y, lx, bl.y);
    r.y = fmaf(b - t, ly, t);
    t = fmaf(tr.z - tl.z, lx, tl.z); b = fmaf(br.z - bl.z, lx, bl.z);
    r.z = fmaf(b - t, ly, t);
    t = fmaf(tr.w - tl.w, lx, tl.w); b = fmaf(br.w - bl.w, lx, bl.w);
    r.w = fmaf(b - t, ly, t);
    store_nt4(o + c0 + 4 * h, r);
  }
}

__global__ __launch_bounds__(BLOCK_THREADS) void pyramid_roi_align_kernel(
    const float* __restrict__ boxes, const float* __restrict__ meta,
    const float* __restrict__ p2, const float* __restrict__ p3,
    const float* __restrict__ p4, const float* __restrict__ p5,
    float* __restrict__ out,
    int N, int H2, int H3, int H4, int H5, int NPIX) {
  // 4 waves * 2 buffers * (4 corners * 256 ch) = 32 KB LDS per block
  __shared__ float lds[WAVES_PER_BLOCK * 2 * 4 * CCH];
  const int lane    = threadIdx.x & 31;
  const int wave    = threadIdx.x >> 5;
  const int waveGid = blockIdx.x * WAVES_PER_BLOCK + wave;
  const int nWaves  = gridDim.x * WAVES_PER_BLOCK;
  float* myLds      = &lds[wave * (2 * 4 * CCH)];
  const int c0      = lane * CHUNK;

  // image_meta[0, 4:7] = (1024,1024,3): scale = sqrt(area)/224
  const float scale = sqrtf(meta[4] * meta[5]) * (1.0f / 224.0f);

  int       havePrev = 0, pvalid = 0, pparity = 0, parity = 0;
  float     plx = 0.f, ply = 0.f;
  long long poff = 0;

  for (int pix = waveGid; pix < NPIX; pix += nWaves) {
    const int box = pix / 49;
    const int rem = pix - box * 49;
    const int py  = rem / POOL;
    const int px  = rem - py * POOL;
    const int b   = box / N;

    const float by1 = boxes[box * 4 + 0];
    const float bx1 = boxes[box * 4 + 1];
    const float by2 = boxes[box * 4 + 2];
    const float bx2 = boxes[box * 4 + 3];
    const float bh = by2 - by1, bw = bx2 - bx1;

    // roi_level = clip(4 + round(log2(sqrt(h*w)*sqrt(area)/224)), 2, 5)
    float lf = logf(sqrtf(bh * bw) * scale) * 1.44269504088896340736f;
    float Lf = 4.0f + rintf(lf);
    Lf = fminf(fmaxf(Lf, 2.0f), 5.0f);
    const int L = (int)Lf;
    const float* fm; int H;
    if      (L == 2) { fm = p2; H = H2; }
    else if (L == 3) { fm = p3; H = H3; }
    else if (L == 4) { fm = p4; H = H4; }
    else             { fm = p5; H = H5; }
    const int   W   = H;
    const float Hm1 = (float)(H - 1), Wm1 = (float)(W - 1);

    const float gy = (float)py / 6.0f;
    const float gx = (float)px / 6.0f;
    const float ys = by1 * Hm1 + gy * (bh * Hm1);
    const float xs = bx1 * Wm1 + gx * (bw * Wm1);
    const int valid = (ys >= 0.f) && (ys <= Hm1) && (xs >= 0.f) && (xs <= Wm1);

    const float y0f = floorf(ys), x0f = floorf(xs);
    const float ly = ys - y0f, lx = xs - x0f;
    int y0  = (int)y0f; y0 = y0 < 0 ? 0 : (y0 > H - 1 ? H - 1 : y0);
    int y1i = y0 + 1 > H - 1 ? H - 1 : y0 + 1;
    int x0  = (int)x0f; x0 = x0 < 0 ? 0 : (x0 > W - 1 ? W - 1 : x0);
    int x1i = x0 + 1 > W - 1 ? W - 1 : x0 + 1;

    const size_t rowC  = (size_t)W * CCH;
    const float* basep = fm + (size_t)b * H * rowC;
    const float* g00 = basep + (size_t)y0  * rowC + (size_t)x0  * CCH + c0;
    const float* g01 = basep + (size_t)y0  * rowC + (size_t)x1i * CCH + c0;
    const float* g10 = basep + (size_t)y1i * rowC + (size_t)x0  * CCH + c0;
    const float* g11 = basep + (size_t)y1i * rowC + (size_t)x1i * CCH + c0;

    // stage 4 corners (4 KB) into this wave's LDS buffer: 8x async b128
    float* dst = myLds + parity * (4 * CCH) + c0;
    async_copy_b128(g00,     dst);
    async_copy_b128(g00 + 4, dst + 4);
    async_copy_b128(g01,     dst + 1 * CCH);
    async_copy_b128(g01 + 4, dst + 1 * CCH + 4);
    async_copy_b128(g10,     dst + 2 * CCH);
    async_copy_b128(g10 + 4, dst + 2 * CCH + 4);
    async_copy_b128(g11,     dst + 3 * CCH);
    async_copy_b128(g11 + 4, dst + 3 * CCH + 4);

    if (havePrev) {
      WAIT_ASYNC(8);  // previous pixel's 8 copies retired; current 8 in flight
      blend_store(myLds + pparity * (4 * CCH), out + poff, plx, ply, pvalid, c0);
    }
    havePrev = 1;
    plx = lx; ply = ly; pvalid = valid;
    poff = (long long)pix * CCH;
    pparity = parity;
    parity ^= 1;
  }
  if (havePrev) {
    WAIT_ASYNC(0);
    blend_store(myLds + pparity * (4 * CCH), out + poff, plx, ply, pvalid, c0);
  }
}

extern "C" void kernel_launch(void* const* d_in, const int* in_sizes, int n_in,
                              void* d_out, int out_size, void* d_ws,
                              size_t ws_size, hipStream_t stream) {
  const float* boxes = (const float*)d_in[0];
  const float* meta  = (const float*)d_in[1];
  const float* p2    = (const float*)d_in[2];
  const float* p3    = (const float*)d_in[3];
  const float* p4    = (const float*)d_in[4];
  const float* p5    = (const float*)d_in[5];
  float* out = (float*)d_out;

  const int B = in_sizes[1] / 20;               // image_meta is [B,20]
  const int N = in_sizes[0] / (4 * B);          // boxes is [B,N,4]
  auto dimOf = [&](int sz) {
    return (int)(std::sqrt((double)sz / ((double)B * CCH)) + 0.5);
  };
  const int H2 = dimOf(in_sizes[2]);
  const int H3 = dimOf(in_sizes[3]);
  const int H4 = dimOf(in_sizes[4]);
  const int H5 = dimOf(in_sizes[5]);

  const int NPIX = B * N * POOL * POOL;
  int blocks = (NPIX + WAVES_PER_BLOCK - 1) / WAVES_PER_BLOCK;
  if (blocks > 2048) blocks = 2048;  // persistent waves, ~12 pixels each

  pyramid_roi_align_kernel<<<blocks, BLOCK_THREADS, 0, stream>>>(
      boxes, meta, p2, p3, p4, p5, out, N, H2, H3, H4, H5, NPIX);
}